// MaskedAttentionHead_81956565942655
// MI455X (gfx1250) — compile-verified
//
#include <hip/hip_runtime.h>
#include <math.h>

typedef float v2f __attribute__((ext_vector_type(2)));
typedef float v8f __attribute__((ext_vector_type(8)));

#define EMBD 768
#define HEAD 64
#define SEQ  4096
#define BATCH 4
#define BS   (BATCH * SEQ)

// D = A(16x4 f32) * B(4x16 f32) + C(16x16 f32)
__device__ __forceinline__ v8f wmma_f32(v2f a, v2f b, v8f c) {
    return __builtin_amdgcn_wmma_f32_16x16x4_f32(
        /*neg_a=*/false, a, /*neg_b=*/false, b,
        /*c_mod=*/(short)0, c, /*reuse_a=*/false, /*reuse_b=*/false);
}

// ---------------------------------------------------------------------------
// Kernel 1: QKV projection.  x[BS,768] @ W[768,64] -> q/k row-major [BS,64],
// v stored transposed vT[B,64,S] so the attention kernel gets contiguous
// b64 fragment loads for the V^T A-matrix.
// One wave per (16-row tile, matrix).  g in [0,3072).
// ---------------------------------------------------------------------------
__global__ __launch_bounds__(256) void qkv_proj_kernel(
    const float* __restrict__ x,
    const float* __restrict__ Wq, const float* __restrict__ Wk,
    const float* __restrict__ Wv,
    float* __restrict__ qb, float* __restrict__ kb, float* __restrict__ vT)
{
    const int lane = threadIdx.x & 31;
    const int wave = threadIdx.x >> 5;
    const int g    = blockIdx.x * 8 + wave;     // 0..3071
    const int mat  = g >> 10;                   // 0=Q 1=K 2=V
    const int tile = g & 1023;                  // 0..1023
    const int row0 = tile << 4;                 // flat row in [0, BS)
    const float* W = (mat == 0) ? Wq : (mat == 1) ? Wk : Wv;

    const int half = lane >> 4;                 // half-wave id
    const int lx   = lane & 15;

    v8f acc[4] = {};                            // 16 rows x 64 cols (4 N-tiles)
    const float* xrow = x + (size_t)(row0 + lx) * EMBD;
    for (int c = 0; c < EMBD; c += 4) {
        const int kk = c + 2 * half;            // A layout: lanes16-31 take K+2
        v2f a = *(const v2f*)(xrow + kk);       // x[m][kk], x[m][kk+1]
        #pragma unroll
        for (int t = 0; t < 4; ++t) {
            v2f bf;                              // B layout: lane = N column
            bf.x = W[(size_t)kk       * HEAD + t * 16 + lx];
            bf.y = W[(size_t)(kk + 1) * HEAD + t * 16 + lx];
            acc[t] = wmma_f32(a, bf, acc[t]);
        }
    }

    if (mat < 2) {                               // Q,K: row-major store
        float* outb = (mat == 0) ? qb : kb;
        #pragma unroll
        for (int t = 0; t < 4; ++t)
            #pragma unroll
            for (int i = 0; i < 8; ++i) {
                const int r = row0 + i + 8 * half;      // C layout row
                outb[(size_t)r * HEAD + t * 16 + lx] = acc[t][i];
            }
    } else {                                     // V: transposed store vT[b][h][s]
        const int batch = row0 >> 12;            // /4096
        const int sbase = row0 & 4095;
        float* vb = vT + (size_t)batch * HEAD * SEQ;
        #pragma unroll
        for (int t = 0; t < 4; ++t)
            #pragma unroll
            for (int i = 0; i < 8; ++i) {
                const int s = sbase + i + 8 * half;
                const int h = t * 16 + lx;
                vb[(size_t)h * SEQ + s] = acc[t][i];
            }
    }
}

// ---------------------------------------------------------------------------
// Kernel 2: causal flash attention, one wave per 16-query tile.
// Computes S^T = K * Q^T (M=key, N=query) so query stays in the lane dim:
// per-lane scalar online-softmax state, xor-16 shuffles only (no LDS).
// out^T = V^T * P^T accumulated in C layout (M=head, N=query).
// ---------------------------------------------------------------------------
__global__ __launch_bounds__(256) void flash_attn_kernel(
    const float* __restrict__ qb, const float* __restrict__ kb,
    const float* __restrict__ vT, float* __restrict__ out)
{
    const int lane  = threadIdx.x & 31;
    const int wave  = threadIdx.x >> 5;
    const int qt    = blockIdx.x * 8 + wave;    // 0..1023
    const int batch = qt >> 8;
    const int q0    = (qt & 255) << 4;
    const int half  = lane >> 4;
    const int lx    = lane & 15;

    const float scale = 0.125f;                 // 1/sqrt(64), folded into Q

    // Q tile as B-matrix fragments (lane = query, VGPRs = head-dim chunks)
    v2f qf[16];
    {
        const float* qrow = qb + (size_t)(batch * SEQ + q0 + lx) * HEAD;
        #pragma unroll
        for (int c = 0; c < 16; ++c) {
            v2f v = *(const v2f*)(qrow + 4 * c + 2 * half);
            qf[c].x = v.x * scale;
            qf[c].y = v.y * scale;
        }
    }

    v8f acc[4] = {};                            // out^T, 4 head M-tiles
    float m = -INFINITY, l = 0.0f;
    const int q_lane = q0 + lx;                 // this lane's query
    const float* kbase = kb + (size_t)batch * SEQ * HEAD;
    const float* vbase = vT + (size_t)batch * HEAD * SEQ;

    const int nkt = (q0 >> 4) + 1;              // causal key-tile count
    for (int kt = 0; kt < nkt; ++kt) {
        const int k0 = kt << 4;

        // S^T tile: A = K[16 keys x 64], B = Q^T[64 x 16 queries]
        v8f st = {};
        const float* krow = kbase + (size_t)(k0 + lx) * HEAD;
        #pragma unroll
        for (int c = 0; c < 16; ++c) {
            v2f kf = *(const v2f*)(krow + 4 * c + 2 * half);
            st = wmma_f32(kf, qf[c], st);
        }

        // causal mask + per-query online softmax (query = lane)
        float p[8];
        float tmax = -INFINITY;
        #pragma unroll
        for (int i = 0; i < 8; ++i) {
            const int key = k0 + i + 8 * half;  // C layout: VGPR row = key
            const float s = (key <= q_lane) ? st[i] : -INFINITY;
            p[i] = s;
            tmax = fmaxf(tmax, s);
        }
        tmax = fmaxf(tmax, __shfl_xor(tmax, 16, 32));
        const float m_new = fmaxf(m, tmax);
        const float alpha = __expf(m - m_new);
        float rsum = 0.0f;
        #pragma unroll
        for (int i = 0; i < 8; ++i) {
            p[i] = __expf(p[i] - m_new);
            rsum += p[i];
        }
        l = l * alpha + rsum + __shfl_xor(rsum, 16, 32);
        m = m_new;
        #pragma unroll
        for (int t = 0; t < 4; ++t)
            #pragma unroll
            for (int i = 0; i < 8; ++i) acc[t][i] *= alpha;

        // P^T B-fragments: own half holds keys 8*half..8*half+7, grab the
        // other 8 via xor-16 shuffles (no LDS round-trip).
        float sh[8];
        #pragma unroll
        for (int i = 0; i < 8; ++i) sh[i] = __shfl_xor(p[i], 16, 32);
        v2f pf[4];
        #pragma unroll
        for (int c = 0; c < 4; ++c) {
            const int keyA = 4 * c + 2 * half;  // key supplied in V0 by this lane
            const int keyB = keyA + 1;
            pf[c].x = ((keyA >> 3) == half) ? p[keyA & 7] : sh[keyA & 7];
            pf[c].y = ((keyB >> 3) == half) ? p[keyB & 7] : sh[keyB & 7];
        }

        // out^T += V^T(16 heads x 16 keys per M-tile) * P^T(16 keys x 16 q)
        #pragma unroll
        for (int t = 0; t < 4; ++t) {
            const float* vrow = vbase + (size_t)(t * 16 + lx) * SEQ + k0;
            #pragma unroll
            for (int c = 0; c < 4; ++c) {
                v2f vf = *(const v2f*)(vrow + 4 * c + 2 * half);
                acc[t] = wmma_f32(vf, pf[c], acc[t]);
            }
        }
    }

    // normalize and store: lane holds query q0+lx, heads t*16 + 8*half + i
    const float inv = 1.0f / l;
    float* orow = out + (size_t)(batch * SEQ + q0 + lx) * HEAD + 8 * half;
    #pragma unroll
    for (int t = 0; t < 4; ++t) {
        float4 lo = make_float4(acc[t][0] * inv, acc[t][1] * inv,
                                acc[t][2] * inv, acc[t][3] * inv);
        float4 hi = make_float4(acc[t][4] * inv, acc[t][5] * inv,
                                acc[t][6] * inv, acc[t][7] * inv);
        *(float4*)(orow + t * 16)     = lo;
        *(float4*)(orow + t * 16 + 4) = hi;
    }
}

extern "C" void kernel_launch(void* const* d_in, const int* in_sizes, int n_in,
                              void* d_out, int out_size, void* d_ws, size_t ws_size,
                              hipStream_t stream) {
    const float* x  = (const float*)d_in[0];
    const float* Wq = (const float*)d_in[1];
    const float* Wk = (const float*)d_in[2];
    const float* Wv = (const float*)d_in[3];

    float* qb = (float*)d_ws;                       // [BS, 64]
    float* kb = qb + (size_t)BS * HEAD;             // [BS, 64]
    float* vT = kb + (size_t)BS * HEAD;             // [B, 64, S]

    // 3072 waves: 3 matrices x 1024 16-row tiles, 8 waves/block
    qkv_proj_kernel<<<384, 256, 0, stream>>>(x, Wq, Wk, Wv, qb, kb, vT);
    // 1024 waves: 4 batches x 256 query tiles, 8 waves/block
    flash_attn_kernel<<<128, 256, 0, stream>>>(qb, kb, vT, (float*)d_out);
}